// BalRNN_90933047591058
// MI455X (gfx1250) — compile-verified
//
#include <hip/hip_runtime.h>

#define HIDDEN   20000
#define KIN      10
#define NPB      256          // neurons per block (8 waves x 32)
#define HK       (HIDDEN*KIN) // 200000
#define H32      (HIDDEN*32)  // 640000
#define SEQH     (100*HIDDEN) // out stride per batch
#define FFC      3.16227766016837933f  // sqrt(10)

typedef __attribute__((ext_vector_type(4))) int v4i_t;

// ---------------- CDNA5 async global->LDS helpers ----------------
__device__ __forceinline__ void async_ld_b128(const void* g, void* l) {
#if defined(__has_builtin) && __has_builtin(__builtin_amdgcn_global_load_async_to_lds_b128)
  __builtin_amdgcn_global_load_async_to_lds_b128(
      (__attribute__((address_space(1))) v4i_t*)g,
      (__attribute__((address_space(3))) v4i_t*)l, 0, 0);
#else
  unsigned lo = (unsigned)(unsigned long long)l;  // low 32 bits = LDS byte offset
  asm volatile("global_load_async_to_lds_b128 %0, %1, off"
               :: "v"(lo), "v"(g) : "memory");
#endif
}

__device__ __forceinline__ void wait_async0() {
#if defined(__has_builtin) && __has_builtin(__builtin_amdgcn_s_wait_asynccnt)
  __builtin_amdgcn_s_wait_asynccnt(0);
#else
  asm volatile("s_wait_asynccnt 0" ::: "memory");
#endif
}

// Stage rem*KIN ints + rem*KIN floats into LDS (rem*KIN divisible by 4).
__device__ __forceinline__ void stage_idx_val(const int* gi, const float* gv,
                                              int* si, float* sv,
                                              int rem, int tid, int nthreads) {
  int nchunk = (rem * KIN) >> 2;
  for (int c = tid; c < nchunk; c += nthreads) {
    async_ld_b128(gi + c * 4, si + c * 4);
    async_ld_b128(gv + c * 4, sv + c * 4);
  }
  wait_async0();
  __syncthreads();
}

// ---------------- zero init of t=0 state ----------------
__global__ __launch_bounds__(256) void zero2_kernel(float* __restrict__ a,
                                                    float* __restrict__ b) {
  int i = blockIdx.x * 256 + threadIdx.x;   // 160000 float4 per array
  float4 z = {0.f, 0.f, 0.f, 0.f};
  ((float4*)a)[i] = z;
  ((float4*)b)[i] = z;
}

// ---------------- layer 0: h0_new = relu(ff + W0 @ h0_prev) ----------------
__global__ __launch_bounds__(256) void layer0_kernel(
    const int* __restrict__ gidx, const float* __restrict__ gval,
    const float* __restrict__ hprev, float* __restrict__ hnext) {
  __shared__ int   s_idx[NPB * KIN];
  __shared__ float s_val[NPB * KIN];
  const int tid   = threadIdx.x;
  const int nbase = blockIdx.x * NPB;
  int rem = HIDDEN - nbase; if (rem > NPB) rem = NPB;   // always multiple of 32
  stage_idx_val(gidx + nbase * KIN, gval + nbase * KIN, s_idx, s_val, rem, tid, 256);

  const int lane = tid & 31;
  const int wv   = tid >> 5;
  for (int j = 0; j < 32; ++j) {
    const int nl = wv * 32 + j;
    const int i  = nbase + nl;
    if (i >= HIDDEN) break;                 // wave-uniform
    float acc = FFC;
#pragma unroll
    for (int k = 0; k < KIN; ++k) {
      const int   ix = s_idx[nl * KIN + k];
      const float v  = s_val[nl * KIN + k];
      acc += v * hprev[ix * 32 + lane];     // coalesced 128B gather
    }
    acc = acc > 0.f ? acc : 0.f;
    hnext[i * 32 + lane] = acc;
  }
}

// ---- layer 1: h1_new = relu(W1 @ h0_new + W1 @ h1_prev); writes out[:,t,:] ----
__global__ __launch_bounds__(256) void layer1_kernel(
    const int* __restrict__ gidx, const float* __restrict__ gval,
    const float* __restrict__ h0new, const float* __restrict__ h1prev,
    float* __restrict__ h1next, float* __restrict__ out, int t) {
  __shared__ int   s_idx[NPB * KIN];
  __shared__ float s_val[NPB * KIN];
  __shared__ float s_tile[8][32 * 33];      // per-wave 32x32 transpose tile (pad 33)
  const int tid   = threadIdx.x;
  const int nbase = blockIdx.x * NPB;
  int rem = HIDDEN - nbase; if (rem > NPB) rem = NPB;
  stage_idx_val(gidx + nbase * KIN, gval + nbase * KIN, s_idx, s_val, rem, tid, 256);

  const int lane = tid & 31;
  const int wv   = tid >> 5;
  if (nbase + wv * 32 >= HIDDEN) return;    // whole-wave guard (HIDDEN % 32 == 0)

  for (int j = 0; j < 32; ++j) {
    const int nl = wv * 32 + j;
    const int i  = nbase + nl;
    float acc = 0.f;
#pragma unroll
    for (int k = 0; k < KIN; ++k) {
      const int   ix = s_idx[nl * KIN + k];
      const float v  = s_val[nl * KIN + k];
      acc += v * (h0new[ix * 32 + lane] + h1prev[ix * 32 + lane]);
    }
    acc = acc > 0.f ? acc : 0.f;
    h1next[i * 32 + lane] = acc;            // coalesced state store
    s_tile[wv][j * 33 + lane] = acc;        // row = neuron, col = batch
  }
  // transpose write: out[b, t, i] with i = nbase + wv*32 + lane (coalesced in lane)
  const int i0 = nbase + wv * 32;
  for (int b = 0; b < 32; ++b)
    out[b * SEQH + t * HIDDEN + i0 + lane] = s_tile[wv][lane * 33 + b];
}

// ---------------- h_final: [2][32][HIDDEN] from transposed state ----------------
__global__ __launch_bounds__(256) void final_kernel(const float* __restrict__ h0,
                                                    const float* __restrict__ h1,
                                                    float* __restrict__ hfin) {
  __shared__ float s_tile[8][32 * 33];
  const int lane = threadIdx.x & 31;
  const int wv   = threadIdx.x >> 5;
  const int tile = blockIdx.x * 8 + wv;     // 1250 tiles = 2 layers * 625
  if (tile >= 2 * (HIDDEN / 32)) return;
  const int l  = tile / (HIDDEN / 32);
  const int i0 = (tile % (HIDDEN / 32)) * 32;
  const float* src = l ? h1 : h0;
  for (int j = 0; j < 32; ++j)
    s_tile[wv][j * 33 + lane] = src[(i0 + j) * 32 + lane];
  for (int b = 0; b < 32; ++b)
    hfin[l * H32 + b * HIDDEN + i0 + lane] = s_tile[wv][lane * 33 + b];
}

extern "C" void kernel_launch(void* const* d_in, const int* in_sizes, int n_in,
                              void* d_out, int out_size, void* d_ws, size_t ws_size,
                              hipStream_t stream) {
  (void)in_sizes; (void)n_in; (void)out_size; (void)ws_size;
  const int*   hh_idx = (const int*)d_in[1];   // [2][HIDDEN][KIN]
  const float* hh_val = (const float*)d_in[2]; // [2][HIDDEN][KIN]
  float* out = (float*)d_out;                  // [32][100][HIDDEN] ++ [2][32][HIDDEN]
  float* w   = (float*)d_ws;
  float* h0[2] = { w,            w + H32     };
  float* h1[2] = { w + 2 * H32,  w + 3 * H32 };

  const int nblocks = (HIDDEN + NPB - 1) / NPB;   // 79

  zero2_kernel<<<H32 / (256 * 4), 256, 0, stream>>>(h0[0], h1[0]);
  for (int t = 0; t < 100; ++t) {
    const int p = t & 1, q = p ^ 1;
    layer0_kernel<<<nblocks, 256, 0, stream>>>(hh_idx, hh_val, h0[p], h0[q]);
    layer1_kernel<<<nblocks, 256, 0, stream>>>(hh_idx + HK, hh_val + HK,
                                               h0[q], h1[p], h1[q], out, t);
  }
  // after t=99, fresh state sits in buffer index 0 of each layer
  final_kernel<<<(2 * (HIDDEN / 32) + 7) / 8, 256, 0, stream>>>(
      h0[0], h1[0], out + 32 * SEQH);
}